// DenseCapsBayesian_20246475833893
// MI455X (gfx1250) — compile-verified
//
#include <hip/hip_runtime.h>
#include <math.h>

// CapsNet DigitCaps dims
#define BB 256
#define II 1152
#define EE 8
#define JJ 10
#define DD 16
#define FLATW (II * JJ * DD * EE)         // 1,474,560
#define PREDS_ELEMS ((size_t)BB * II * JJ * DD)  // 47,185,920 (188.7 MB fp32)

typedef __attribute__((ext_vector_type(2))) float v2f;
typedef __attribute__((ext_vector_type(8))) float v8f;

// tfp softplus shift: log(expm1(1.0))
#define C_SOFTPLUS 0.5413248546129181f
#define KERAS_EPS 1e-7f

// ---------------------------------------------------------------------------
// Kernel 1: Bayesian weight sample  W = loc + (1e-5 + softplus(c + rs)) * eps
// ---------------------------------------------------------------------------
__global__ void caps_build_w(const float* __restrict__ loc,
                             const float* __restrict__ rs,
                             const float* __restrict__ eps,
                             float* __restrict__ W, int n) {
    int t = blockIdx.x * blockDim.x + threadIdx.x;
    if (t < n) {
        float x = C_SOFTPLUS + rs[t];
        // numerically stable softplus
        float sp = (x > 20.0f) ? x : log1pf(expf(x));
        W[t] = loc[t] + (1e-5f + sp) * eps[t];
    }
}

// ---------------------------------------------------------------------------
// Kernel 2: preds[b,i,j,d] = sum_e W[i,j,d,e] * inputs[b,i,e]
// One wave per (i, j). M = batch (16 tiles of 16), N = d (exactly 16), K = 8
// (two V_WMMA_F32_16X16X4_F32 steps). W tile (B operand) loaded once per
// wave and reused across all 16 batch tiles.
// ---------------------------------------------------------------------------
__global__ void caps_preds_wmma(const float* __restrict__ inputs,  // [B,I,E]
                                const float* __restrict__ W,       // [I,J,D,E]
                                float* __restrict__ preds) {       // [B,I,J,D]
    int wave = (blockIdx.x * blockDim.x + threadIdx.x) >> 5;  // wave32
    int lane = threadIdx.x & 31;
    if (wave >= II * JJ) return;  // wave-uniform (grid is exact anyway)

    const int i = wave / JJ;
    const int j = wave % JJ;
    const int n = lane & 15;        // column (= d) for B and D operands
    const int half = lane >> 4;     // K-half selector for A/B operands

    // B operand: B[k,n] = W[i, j*16 + n, k]; lane holds K = 2*half + {0,1}
    const float* wp = W + (size_t)i * (JJ * DD * EE) + (size_t)(j * DD + n) * EE;
    v2f b0 = *(const v2f*)(wp + 2 * half);       // K = 0..3
    v2f b1 = *(const v2f*)(wp + 4 + 2 * half);   // K = 4..7

    const size_t preds_row_stride = (size_t)II * JJ * DD;  // stride over b
    float* dp = preds + (size_t)i * (JJ * DD) + (size_t)j * DD + n;

    for (int mt = 0; mt < 16; ++mt) {
        // A operand: A[m,k] = inputs[b0+m, i, k]; row m = lane%16,
        // lane holds K = 2*half + {0,1} per VGPR pair
        const float* ap = inputs + (size_t)(mt * 16 + n) * (II * EE) + (size_t)i * EE;
        v2f a0 = *(const v2f*)(ap + 2 * half);
        v2f a1 = *(const v2f*)(ap + 4 + 2 * half);

        v8f c = {};
        c = __builtin_amdgcn_wmma_f32_16x16x4_f32(false, a0, false, b0,
                                                  (short)0, c, false, false);
        c = __builtin_amdgcn_wmma_f32_16x16x4_f32(false, a1, false, b1,
                                                  (short)0, c, false, false);

        // D layout: lane l, VGPR v -> row (v + 8*half), col l%16
#pragma unroll
        for (int v = 0; v < 8; ++v) {
            int row = mt * 16 + v + 8 * half;  // batch index
            dp[(size_t)row * preds_row_stride] = c[v];
        }
    }
}

// ---------------------------------------------------------------------------
// Kernel 3: fused dynamic routing (3 iterations), one block per batch elem.
// raw[I][J] + softmax stats + out[J][D] live in LDS (~56 KB).
// ---------------------------------------------------------------------------
__global__ void caps_routing(const float* __restrict__ preds,  // [B,I,J,D]
                             float* __restrict__ out,          // [B,J,D]
                             float* __restrict__ rw_out) {     // [B,I,J]
    __shared__ float raw_s[II * JJ];   // 46080 B
    __shared__ float mx_s[II];         // 4608 B
    __shared__ float dn_s[II];         // 4608 B
    __shared__ float out_s[JJ * DD];   // 640 B
    __shared__ float fac_s[JJ];        // 40 B

    const int b = blockIdx.x;
    const int t = threadIdx.x;
    const int NT = blockDim.x;  // 256
    const float* pb = preds + (size_t)b * II * JJ * DD;

    for (int idx = t; idx < II * JJ; idx += NT) raw_s[idx] = 0.0f;
    __syncthreads();

    for (int it = 0; it < 3; ++it) {
        // --- softmax stats over J per input capsule i ---
        for (int i = t; i < II; i += NT) {
            float mx = -INFINITY;
#pragma unroll
            for (int j = 0; j < JJ; ++j) mx = fmaxf(mx, raw_s[i * JJ + j]);
            float dn = 0.0f;
#pragma unroll
            for (int j = 0; j < JJ; ++j) dn += expf(raw_s[i * JJ + j] - mx);
            mx_s[i] = mx;
            dn_s[i] = dn;
        }
        __syncthreads();

        // --- out[j,d] = sum_i rw[i,j] * preds[b,i,j,d] ---
        if (t < JJ * DD) {
            const int j = t / DD, d = t % DD;
            float acc = 0.0f;
            for (int i = 0; i < II; ++i) {
                float rw = expf(raw_s[i * JJ + j] - mx_s[i]) / dn_s[i];
                acc += rw * pb[(size_t)i * (JJ * DD) + j * DD + d];
            }
            out_s[t] = acc;
        }
        __syncthreads();

        // --- squash over capsule dim D ---
        if (t < JJ) {
            float s2 = 0.0f;
#pragma unroll
            for (int d = 0; d < DD; ++d) {
                float v = out_s[t * DD + d];
                s2 += v * v;
            }
            fac_s[t] = (s2 / (1.0f + s2)) / sqrtf(s2 + KERAS_EPS);
        }
        __syncthreads();
        if (t < JJ * DD) out_s[t] *= fac_s[t / DD];
        __syncthreads();

        // --- agreement: raw[i,j] += <preds[b,i,j,:], out[j,:]> ---
        if (it < 2) {
            for (int idx = t; idx < II * JJ; idx += NT) {
                const int i = idx / JJ, j = idx % JJ;
                const float* pp = pb + (size_t)i * (JJ * DD) + j * DD;
                float dot = 0.0f;
#pragma unroll
                for (int d = 0; d < DD; ++d) dot += pp[d] * out_s[j * DD + d];
                raw_s[idx] += dot;
            }
            __syncthreads();
        }
    }

    // --- write outputs: out [B,J,D] then rw (= softmax of final raw) [B,I,J]
    if (t < JJ * DD) out[(size_t)b * JJ * DD + t] = out_s[t];
    for (int idx = t; idx < II * JJ; idx += NT) {
        const int i = idx / JJ;
        rw_out[(size_t)b * II * JJ + idx] = expf(raw_s[idx] - mx_s[i]) / dn_s[i];
    }
}

// ---------------------------------------------------------------------------
extern "C" void kernel_launch(void* const* d_in, const int* in_sizes, int n_in,
                              void* d_out, int out_size, void* d_ws, size_t ws_size,
                              hipStream_t stream) {
    const float* inputs    = (const float*)d_in[0];  // [B,I,E]
    const float* loc       = (const float*)d_in[1];  // [FLATW]
    const float* raw_scale = (const float*)d_in[2];  // [FLATW]
    const float* eps       = (const float*)d_in[3];  // [FLATW]

    float* out_p  = (float*)d_out;                 // [B,J,D] = 40960 floats
    float* rw_p   = out_p + (size_t)BB * JJ * DD;  // [B,I,J] = 2,949,120 floats

    // workspace: W then preds
    float* W     = (float*)d_ws;           // 1,474,560 floats (5.9 MB)
    float* preds = W + FLATW;              // 47,185,920 floats (188.7 MB, L2-resident)

    // 1) weight sample
    caps_build_w<<<(FLATW + 255) / 256, 256, 0, stream>>>(loc, raw_scale, eps, W, FLATW);

    // 2) predictions via WMMA: II*JJ = 11520 waves, 8 waves/block -> 1440 blocks
    caps_preds_wmma<<<(II * JJ) / 8, 256, 0, stream>>>(inputs, W, preds);

    // 3) fused routing, one block per batch element
    caps_routing<<<BB, 256, 0, stream>>>(preds, out_p, rw_p);
}